// MultiBoxEval_67937792688195
// MI455X (gfx1250) — compile-verified
//
#include <hip/hip_runtime.h>
#include <stdint.h>

#define NB 64      // batch
#define NP 5000    // preds per image
#define NG 300     // targets per image
#define SCALE 512.0f
#define THR 0.5f

// ---------------------------------------------------------------------------
// Phase 1: fully parallel per-pred work.
// For each pred: IoU vs all 300 targets (staged to LDS via CDNA5 async copy),
// class-masked argmax, candidate test. Packs {flag,bi} into stats slot 0 (as
// raw int bits), writes score/cls into slots 1/2.
// ---------------------------------------------------------------------------
__global__ __launch_bounds__(256)
void phase1_kernel(const float* __restrict__ output,
                   const float* __restrict__ labels,
                   float* __restrict__ out) {
    __shared__ float s_t[NG * 5];   // scaled target boxes + cls
    __shared__ float s_area[NG];    // target areas

    const int b   = blockIdx.y;
    const int tid = threadIdx.x;

    // ---- async global -> LDS stage of this image's 300x5 target table ----
    const float* lab = labels + (size_t)b * NG * 5;
    for (int i = tid; i < NG * 5; i += 256) {
        unsigned lds_addr = (unsigned)(uintptr_t)(&s_t[i]);   // low 32b of flat shared ptr == LDS byte addr
        unsigned voff     = (unsigned)(i * 4);
        asm volatile("global_load_async_to_lds_b32 %0, %1, %2 offset:0"
                     :: "v"(lds_addr), "v"(voff), "s"(lab) : "memory");
    }
    asm volatile("s_wait_asynccnt 0" ::: "memory");
    __syncthreads();

    // scale boxes in place, precompute areas
    for (int t = tid; t < NG; t += 256) {
        float x1 = s_t[t*5+0] * SCALE, y1 = s_t[t*5+1] * SCALE;
        float x2 = s_t[t*5+2] * SCALE, y2 = s_t[t*5+3] * SCALE;
        s_t[t*5+0] = x1; s_t[t*5+1] = y1; s_t[t*5+2] = x2; s_t[t*5+3] = y2;
        s_area[t] = (x2 - x1) * (y2 - y1);
    }
    __syncthreads();

    const int p = blockIdx.x * 256 + tid;
    if (p >= NP) return;

    const float* pr = output + ((size_t)b * NP + p) * 6;
    float px1 = pr[0] * SCALE, py1 = pr[1] * SCALE;
    float px2 = pr[2] * SCALE, py2 = pr[3] * SCALE;
    float score = pr[4], pcls = pr[5];
    float parea = (px2 - px1) * (py2 - py1);

    float best = -1.0f;
    int   bi   = 0;
    bool  any  = false;
    for (int t = 0; t < NG; ++t) {
        float tx1 = s_t[t*5+0], ty1 = s_t[t*5+1];
        float tx2 = s_t[t*5+2], ty2 = s_t[t*5+3];
        float lx = fmaxf(px1, tx1), ly = fmaxf(py1, ty1);
        float rx = fminf(px2, tx2), ry = fminf(py2, ty2);
        float w  = fmaxf(rx - lx, 0.0f);
        float h  = fmaxf(ry - ly, 0.0f);
        float inter = w * h;
        float iou = inter / (parea + s_area[t] - inter);   // IEEE div, matches JAX fp32
        bool  cm  = (s_t[t*5+4] == pcls);
        any |= cm;
        float v = cm ? iou : -1.0f;
        if (v > best) { best = v; bi = t; }                // strict > => first-max, jnp.argmax semantics
    }

    unsigned flag = (score > 0.0f) && any && (best > THR);
    unsigned packed = (unsigned)bi | (flag << 31);

    size_t o = ((size_t)b * NP + p) * 3;
    ((unsigned*)out)[o] = packed;   // temp: packed candidate, phase 2 overwrites
    out[o + 1] = score;
    out[o + 2] = pcls;
}

// ---------------------------------------------------------------------------
// Phase 2: one block per image. Async-stage the 5000 packed candidate words
// into LDS, build a wave32 ballot bitmask of candidates, then a tiny serial
// walk over ~157 mask words resolves the detected-mask recurrence exactly.
// Overwrites stats slot 0 with 0.0/1.0 and emits tcls.
// ---------------------------------------------------------------------------
__global__ __launch_bounds__(256)
void phase2_kernel(const float* __restrict__ labels,
                   float* __restrict__ out) {
    __shared__ int           s_cand[NP];          // 20000 B
    __shared__ unsigned      s_mask[160];         // ballot words (157 used)
    __shared__ unsigned char s_corr[NP];
    __shared__ unsigned char s_det[NG];

    const int b   = blockIdx.x;
    const int tid = threadIdx.x;

    // async stage packed ints from stats slot 0 (stride 12 bytes)
    for (int p = tid; p < NP; p += 256) {
        unsigned lds_addr = (unsigned)(uintptr_t)(&s_cand[p]);
        unsigned voff     = (unsigned)((((size_t)b * NP + p) * 3) * 4);
        asm volatile("global_load_async_to_lds_b32 %0, %1, %2 offset:0"
                     :: "v"(lds_addr), "v"(voff), "s"(out) : "memory");
    }
    for (int p = tid; p < NP; p += 256) s_corr[p] = 0;
    for (int t = tid; t < NG; t += 256) s_det[t] = 0;
    asm volatile("s_wait_asynccnt 0" ::: "memory");
    __syncthreads();

    // wave32 ballot compaction: one 32-bit word per 32 preds
    const int lane = tid & 31;
    const int wv   = tid >> 5;                    // 8 waves per block
    for (int base = 0; base < NP; base += 256) {
        int p = base + tid;
        int flag = (p < NP) ? (int)(((unsigned)s_cand[p]) >> 31) : 0;
        unsigned m = (unsigned)__ballot(flag);
        if (lane == 0) s_mask[(base >> 5) + wv] = m;
    }
    __syncthreads();

    // serial recurrence over candidate bits only (order-preserving)
    if (tid == 0) {
        const int NW = (NP + 31) / 32;            // 157
        for (int w = 0; w < NW; ++w) {
            unsigned m = s_mask[w];
            while (m) {
                int l = __ffs(m) - 1;
                m &= m - 1;
                int p  = (w << 5) + l;
                int bi = s_cand[p] & 0x7FFFFFFF;
                if (!s_det[bi]) { s_det[bi] = 1; s_corr[p] = 1; }
            }
        }
    }
    __syncthreads();

    // write final correct flags (overwrite packed temp) + tcls
    size_t ob = (size_t)b * NP * 3;
    for (int p = tid; p < NP; p += 256)
        out[ob + 3 * (size_t)p] = (float)s_corr[p];

    float* tout = out + (size_t)NB * NP * 3 + (size_t)b * NG;
    for (int g = tid; g < NG; g += 256)
        tout[g] = labels[((size_t)b * NG + g) * 5 + 4];
}

extern "C" void kernel_launch(void* const* d_in, const int* in_sizes, int n_in,
                              void* d_out, int out_size, void* d_ws, size_t ws_size,
                              hipStream_t stream) {
    const float* output = (const float*)d_in[0];   // (B,P,6)
    const float* labels = (const float*)d_in[1];   // (B,G,5)
    float* out = (float*)d_out;                    // (B,P,3) stats ++ (B,G) tcls

    dim3 g1((NP + 255) / 256, NB);
    phase1_kernel<<<g1, 256, 0, stream>>>(output, labels, out);
    phase2_kernel<<<NB, 256, 0, stream>>>(labels, out);
}